// VIN_14061722927162
// MI455X (gfx1250) — compile-verified
//
#include <hip/hip_runtime.h>
#include <hip/hip_bf16.h>
#include <math.h>

typedef __attribute__((ext_vector_type(2))) float v2f;
typedef __attribute__((ext_vector_type(8))) float v8f;

#define IMS   64
#define PW    66
#define PAREA (PW*PW)      // 4356 padded cells (= 1089 x 16B, no B128 tail)
#define LI    2
#define LH    150
#define LQ    10
#define BATCH 128

// d_ws float layout: [0..17] fused 2x3x3 reward weights, [18] fused bias,
// [64 ..) padded reward maps rpad[128][66][66]
#define WS_RPAD_OFF 64

// ---------------------------------------------------------------------------
// Kernel 1: fuse h_w (150,2,3,3) + h_b (150) with the 1x1 r_w (150) into a
// single 2-channel 3x3 conv + scalar bias (exact linear fusion).
// ---------------------------------------------------------------------------
__global__ void vin_fuse(const float* __restrict__ hw,
                         const float* __restrict__ hb,
                         const float* __restrict__ rw,
                         float* __restrict__ ws) {
    int j = threadIdx.x;
    if (j < 18) {
        float s = 0.f;
        for (int c = 0; c < LH; ++c) s += rw[c] * hw[c * 18 + j];
        ws[j] = s;
    } else if (j == 18) {
        float s = 0.f;
        for (int c = 0; c < LH; ++c) s += rw[c] * hb[c];
        ws[18] = s;
    }
}

// ---------------------------------------------------------------------------
// Kernel 2: r = conv3x3(input, w_eff, SAME) + b_eff, stored zero-padded as
// rpad[b][66][66] so the iteration kernel never branches on borders.
// ---------------------------------------------------------------------------
__global__ void vin_reward(const float* __restrict__ in,
                           const float* __restrict__ ws,
                           float* __restrict__ rpad) {
    const int b = blockIdx.x;
    float weff[18];
#pragma unroll
    for (int j = 0; j < 18; ++j) weff[j] = ws[j];
    const float beff = ws[18];

    float* rp = rpad + (size_t)b * PAREA;
    for (int i = threadIdx.x; i < PAREA; i += blockDim.x) rp[i] = 0.f;
    __syncthreads();

    const float* inb = in + (size_t)b * LI * IMS * IMS;
    for (int pix = threadIdx.x; pix < IMS * IMS; pix += blockDim.x) {
        int y = pix >> 6, x = pix & 63;
        float acc = beff;
#pragma unroll
        for (int ci = 0; ci < LI; ++ci)
#pragma unroll
            for (int dy = -1; dy <= 1; ++dy)
#pragma unroll
                for (int dx = -1; dx <= 1; ++dx) {
                    int yy = y + dy, xx = x + dx;
                    if (yy >= 0 && yy < IMS && xx >= 0 && xx < IMS)
                        acc += weff[ci * 9 + (dy + 1) * 3 + (dx + 1)] *
                               inb[(ci * IMS + yy) * IMS + xx];
                }
        rp[(y + 1) * PW + (x + 1)] = acc;
    }
}

// ---------------------------------------------------------------------------
// Kernel 3: full value iteration, one workgroup per batch image.
//   - rpad staged into LDS with GLOBAL_LOAD_ASYNC_TO_LDS_B128 (ASYNCcnt DMA)
//   - rpad + double-buffered vpad live in LDS (3*4356 floats = 52 KB)
//   - conv as WMMA f32 16x16x4 GEMM: D[16 act][16 px] = W x V_taps + C_r
//   - C_r (reward conv, iteration-invariant) cached in 64 VGPRs/thread
//   - v starts at 0, loop runs K+1 times (pass 0 yields v0 = max(q_init))
//   - finishes with the gather + FC + softmax for its batch element
// ---------------------------------------------------------------------------
__global__ __launch_bounds__(1024) void vin_iterate(
    const float* __restrict__ rpad_g,
    const float* __restrict__ qw,     // (10,1,3,3)
    const float* __restrict__ wwt,    // (10,1,3,3) value-feedback weights
    const int*   __restrict__ kptr,
    const int*   __restrict__ sx,
    const int*   __restrict__ sy,
    const float* __restrict__ fcw,    // (8,10)
    float*       __restrict__ out)    // [1024 logits | 1024 probs]
{
    __shared__ __align__(16) float lds[3 * PAREA];
    float* rb    = lds;               // padded reward map
    float* vbuf0 = lds + PAREA;       // v ping
    float* vbuf1 = lds + 2 * PAREA;   // v pong

    const int b    = blockIdx.x;
    const int tid  = threadIdx.x;
    const int lane = tid & 31;
    const int p    = lane & 15;       // action row (A/D) and pixel col (B/D)
    const int hi   = (lane >> 4) & 1; // K-half select per WMMA f32 layout
    const int wv   = tid >> 5;        // wave id 0..31

    // Stage reward map via async global->LDS DMA (no VGPR round-trip),
    // while zeroing both v buffers with plain ds stores (borders stay 0).
    const float* rg = rpad_g + (size_t)b * PAREA;
    {
        const unsigned lds_base = (unsigned)(unsigned long long)rb;
        for (int i = tid; i < (PAREA / 4); i += 1024) {       // 1089 x B128
            unsigned long long ga = (unsigned long long)(rg + 4 * i);
            unsigned           la = lds_base + 16u * (unsigned)i;
            asm volatile("global_load_async_to_lds_b128 %0, %1, off"
                         :: "v"(la), "v"(ga) : "memory");
        }
    }
    for (int i = tid; i < PAREA; i += 1024) { vbuf0[i] = 0.f; vbuf1[i] = 0.f; }
    asm volatile("s_wait_asynccnt 0x0" ::: "memory");
    __syncthreads();

    // Per-lane tap geometry + weights for the 3 K-chunks (taps 0..8, pad to 12).
    // A 16x4 layout: lane=M(action), VGPR j / half-wave hi => K = 4c+2*hi+j.
    // B 4x16 layout: lane=N(pixel),  VGPR j / half-wave hi => K = 4c+2*hi+j.
    int   rel[3][2];
    float aq[3][2], aw[3][2];
#pragma unroll
    for (int c = 0; c < 3; ++c)
#pragma unroll
        for (int j = 0; j < 2; ++j) {
            int t  = 4 * c + 2 * hi + j;       // 0..11
            int tt = (t < 9) ? t : 8;          // clamp pad taps to a real addr
            int dy = tt / 3 - 1, dx = tt % 3 - 1;
            rel[c][j] = dy * PW + dx + p;      // per-lane LDS offset
            float q = 0.f, w = 0.f;
            if (p < LQ && t < 9) { q = qw[p * 9 + t]; w = wwt[p * 9 + t]; }
            aq[c][j] = q; aw[c][j] = w;        // zero rows/taps in padding
        }

    // 8 strips of 16 pixels per wave (256 strips cover the 64x64 image)
    int sbase[8];
#pragma unroll
    for (int i = 0; i < 8; ++i) {
        int s = wv * 8 + i;
        int y = s >> 2, x0 = (s & 3) << 4;
        sbase[i] = (y + 1) * PW + (x0 + 1);
    }

    // Iteration-invariant reward contribution: C_r = conv(r, q_w) per strip
    v8f cr[8];
#pragma unroll
    for (int i = 0; i < 8; ++i) {
        v8f acc = {};
#pragma unroll
        for (int c = 0; c < 3; ++c) {
            v2f a;  a[0]  = aq[c][0];                 a[1]  = aq[c][1];
            v2f bb; bb[0] = rb[sbase[i] + rel[c][0]]; bb[1] = rb[sbase[i] + rel[c][1]];
            acc = __builtin_amdgcn_wmma_f32_16x16x4_f32(
                false, a, false, bb, (short)0, acc, false, false);
        }
        cr[i] = acc;
    }

    const int K = *kptr;
    for (int it = 0; it <= K; ++it) {
        const float* vc = (it & 1) ? vbuf1 : vbuf0;
        float*       vn = (it & 1) ? vbuf0 : vbuf1;
#pragma unroll
        for (int i = 0; i < 8; ++i) {
            v8f d = cr[i];
#pragma unroll
            for (int c = 0; c < 3; ++c) {
                v2f a;  a[0]  = aw[c][0];                 a[1]  = aw[c][1];
                v2f bb; bb[0] = vc[sbase[i] + rel[c][0]]; bb[1] = vc[sbase[i] + rel[c][1]];
                d = __builtin_amdgcn_wmma_f32_16x16x4_f32(
                    false, a, false, bb, (short)0, d, false, false);
            }
            // max over 10 actions: lanes<16 hold actions 0-7 in d[0..7],
            // lanes>=16 hold actions 8,9 in d[0..1] (10-15 are zero-weight pad)
            float mlo = fmaxf(fmaxf(fmaxf(d[0], d[1]), fmaxf(d[2], d[3])),
                              fmaxf(fmaxf(d[4], d[5]), fmaxf(d[6], d[7])));
            float mhi = fmaxf(d[0], d[1]);
            float m   = hi ? mhi : mlo;
            m = fmaxf(m, __shfl_xor(m, 16, 32));
            vn[sbase[i] + p] = m;   // lanes p and p+16 write identical value
        }
        __syncthreads();
    }

    // Final q = conv(r,q_w) + conv(v,w) at (state_x, state_y), FC, softmax.
    const float* vf = (K & 1) ? vbuf0 : vbuf1;
    if (tid == 0) {
        int row = sx[b], col = sy[b];
        float q[LQ];
#pragma unroll
        for (int a = 0; a < LQ; ++a) {
            float acc = 0.f;
#pragma unroll
            for (int t = 0; t < 9; ++t) {
                int dy = t / 3 - 1, dx = t % 3 - 1;
                int idx = (row + 1 + dy) * PW + (col + 1 + dx);
                acc += qw[a * 9 + t] * rb[idx] + wwt[a * 9 + t] * vf[idx];
            }
            q[a] = acc;
        }
        float lg[8], mx = -3.4e38f;
#pragma unroll
        for (int j = 0; j < 8; ++j) {
            float s = 0.f;
#pragma unroll
            for (int a = 0; a < LQ; ++a) s += q[a] * fcw[j * LQ + a];
            lg[j] = s; mx = fmaxf(mx, s);
        }
        float ex[8], se = 0.f;
#pragma unroll
        for (int j = 0; j < 8; ++j) { ex[j] = expf(lg[j] - mx); se += ex[j]; }
        float inv = 1.f / se;
#pragma unroll
        for (int j = 0; j < 8; ++j) {
            out[b * 8 + j]             = lg[j];
            out[BATCH * 8 + b * 8 + j] = ex[j] * inv;
        }
    }
}

// ---------------------------------------------------------------------------
extern "C" void kernel_launch(void* const* d_in, const int* in_sizes, int n_in,
                              void* d_out, int out_size, void* d_ws, size_t ws_size,
                              hipStream_t stream) {
    const float* in  = (const float*)d_in[0];  // input_view (128,2,64,64)
    const int*   sx  = (const int*)  d_in[1];  // state_x (128)
    const int*   sy  = (const int*)  d_in[2];  // state_y (128)
    const int*   kp  = (const int*)  d_in[3];  // k (scalar)
    const float* hw  = (const float*)d_in[4];  // h_w (150,2,3,3)
    const float* hb  = (const float*)d_in[5];  // h_b (150)
    const float* rw  = (const float*)d_in[6];  // r_w (1,150,1,1)
    const float* qw  = (const float*)d_in[7];  // q_w (10,1,3,3)
    const float* ww  = (const float*)d_in[8];  // w   (10,1,3,3)
    const float* fcw = (const float*)d_in[9];  // fc_w (8,10)
    float* out  = (float*)d_out;
    float* ws   = (float*)d_ws;
    float* rpad = ws + WS_RPAD_OFF;            // 128*66*66 floats (~2.2 MB)

    vin_fuse   <<<1,     64,   0, stream>>>(hw, hb, rw, ws);
    vin_reward <<<BATCH, 256,  0, stream>>>(in, ws, rpad);
    vin_iterate<<<BATCH, 1024, 0, stream>>>(rpad, qw, ww, kp, sx, sy, fcw, out);
}